// CIoUYOLOLoss_87393994539530
// MI455X (gfx1250) — compile-verified
//
#include <hip/hip_runtime.h>

// ---------------- problem constants (from reference setup_inputs) -----------
#define GRID   76
#define NANCH  3
#define NCLS   20
#define NBATCH 64
#define GG     (GRID * GRID)                  // 5776
#define NCELLS (NBATCH * NANCH * GG)          // 1,108,992
#define CHUNK  256
#define NCHUNKS (NCELLS / CHUNK)              // 4332 (exact)
#define TGT_PER_CHUNK (CHUNK * 5)             // 1280 floats = 5120 B
#define EPSF   1e-7f
#define PI_F   3.14159265358979323846f

static_assert(NCELLS % CHUNK == 0, "chunking must be exact");

typedef __attribute__((ext_vector_type(4))) unsigned int u32x4;
typedef __attribute__((ext_vector_type(4))) int          i32x4;
typedef __attribute__((ext_vector_type(8))) int          i32x8;

__device__ __forceinline__ float softplusf(float x) {
  // log(1 + e^x); -log(sigmoid(-x)) — numerically stable for our range
  return (x > 15.0f) ? x : __logf(1.0f + __expf(x));
}
__device__ __forceinline__ float sigmoidf(float x) {
  return 1.0f / (1.0f + __expf(-x));
}
__device__ __forceinline__ float clamp01(float x) {
  return fminf(fmaxf(x, 0.0f), 1.0f);
}

// CDNA5 TDM: async DMA of `TGT_PER_CHUNK` floats (one chunk's targets) from
// global memory into LDS. dim0 = remaining elements (OOB reads return zero).
__device__ __forceinline__ void tdm_load_targets(const float* gsrc,
                                                 unsigned int ldsAddr,
                                                 unsigned int dim0) {
  unsigned long long ga = (unsigned long long)(const void*)gsrc;

  u32x4 g0;
  g0[0] = 1u;                                              // count=1 (valid), is_restore=0, gather off
  g0[1] = ldsAddr;                                         // lds_addr (bytes)
  g0[2] = (unsigned int)(ga & 0xFFFFFFFFull);              // global_addr[31:0]
  g0[3] = (unsigned int)((ga >> 32) & 0x1FFFFFFull)        // global_addr[56:32]
        | (2u << 30);                                      // type = 2 ("image")

  i32x8 g1 = {0, 0, 0, 0, 0, 0, 0, 0};
  g1[0] = (int)(2u << 16);                                 // data_size = 4 bytes
  g1[1] = (int)((dim0 & 0xFFFFu) << 16);                   // tensor_dim0[15:0]
  g1[2] = (int)((dim0 >> 16) & 0xFFFFu) | (1 << 16);       // tensor_dim0[31:16], tensor_dim1 = 1
  g1[3] = (int)(TGT_PER_CHUNK << 16);                      // tile_dim0 = 1280
  g1[4] = 1;                                               // tile_dim1 = 1 (single line)
  g1[5] = TGT_PER_CHUNK;                                   // tensor_dim0_stride (benign for 1 line)

  i32x4 g2 = {0, 0, 0, 0};
  i32x4 g3 = {0, 0, 0, 0};
  i32x8 g4 = {0, 0, 0, 0, 0, 0, 0, 0};

  // amdgpu-toolchain (clang-23 / therock-10.0) 6-arg form
  __builtin_amdgcn_tensor_load_to_lds(g0, g1, g2, g3, g4, 0);
}

__device__ __forceinline__ unsigned int lds_offset_of(const void* p) {
  // Flat shared-aperture pointers carry the LDS byte offset in addr[31:0].
  return (unsigned int)(unsigned long long)p;
}

// ---------------------------------------------------------------------------
// Pass 1: per-block partial sums of the 5 reduction quantities.
//   q0 = n_obj, q1 = sum obj CIoU loss, q2 = sum obj conf BCE,
//   q3 = sum noobj conf BCE, q4 = sum obj class BCE
// Targets are staged global->LDS with the Tensor Data Mover, double-buffered:
// chunk i+1's DMA is issued before consuming chunk i, and the consumer waits
// with s_wait_tensorcnt(1) (TENSORcnt is per-wave and in-order).
// ---------------------------------------------------------------------------
__global__ void __launch_bounds__(256) yolo_ciou_partials(
    const float* __restrict__ pred,     // (B, 75, G, G)
    const float* __restrict__ tgt,      // (B, A, G, G, 5)
    const float* __restrict__ anchors,  // (A, 2)
    float* __restrict__ ws)             // (gridDim.x, 5) block partials
{
  __shared__ float sTgt[2][TGT_PER_CHUNK]; // ping-pong TDM staging (10240 B)
  __shared__ float sRed[8][5];

  const int tid  = threadIdx.x;
  const float invG = 1.0f / (float)GRID;
  const unsigned int totalT = (unsigned int)NCELLS * 5u;

  float acc_nobj = 0.0f, acc_coord = 0.0f, acc_objb = 0.0f;
  float acc_noobj = 0.0f, acc_cls = 0.0f;

  // Prologue: prefetch this block's first chunk into buffer 0.
  const int chk0 = blockIdx.x;           // grid is clamped to <= NCHUNKS blocks
  if (tid < 32) {
    const unsigned int e0 = (unsigned int)chk0 * (unsigned int)TGT_PER_CHUNK;
    tdm_load_targets(tgt + (size_t)e0, lds_offset_of(&sTgt[0][0]), totalT - e0);
  }

  int bufSel = 0;
  for (int chk = chk0; chk < NCHUNKS; chk += gridDim.x, bufSel ^= 1) {
    const int cell0 = chk * CHUNK;
    const int nxt   = chk + gridDim.x;
    const bool hasNext = nxt < NCHUNKS;

    if (tid < 32) {
      if (hasNext) {
        // Issue next chunk's DMA into the other buffer, then wait until only
        // that newest transfer remains outstanding -> current buffer ready.
        const unsigned int en = (unsigned int)nxt * (unsigned int)TGT_PER_CHUNK;
        tdm_load_targets(tgt + (size_t)en, lds_offset_of(&sTgt[bufSel ^ 1][0]),
                         totalT - en);
        __builtin_amdgcn_s_wait_tensorcnt(1);
      } else {
        __builtin_amdgcn_s_wait_tensorcnt(0);
      }
    }
    __syncthreads();

    // ---- per-cell math (one cell per thread; NCELLS % CHUNK == 0, no tail)
    const int cell = cell0 + tid;
    const int g    = cell % GG;            // gy*G + gx
    const int ba   = cell / GG;            // b*A + a
    const int a    = ba % NANCH;
    const int gx   = g % GRID;
    const int gy   = g / GRID;

    const float* pb = pred + (size_t)ba * 25 * GG + g;   // channel stride = GG
    const float* tl = &sTgt[bufSel][tid * 5];            // conflict-free (gcd(5,64)=1)

    const float p4 = pb[4 * GG];                         // conf logit (always needed)
    const float t4 = tl[4];
    const bool obj = t4 > 0.0f;

    if (obj) {
      acc_nobj += 1.0f;
      acc_objb += softplusf(-p4);                        // -log(sigmoid(p4)), tconf==1

      const float aw = anchors[2 * a]     * invG;
      const float ah = anchors[2 * a + 1] * invG;

      const float p0 = pb[0], p1 = pb[GG], p2 = pb[2 * GG], p3 = pb[3 * GG];
      const float pcx = ((float)gx + sigmoidf(p0)) * invG;
      const float pcy = ((float)gy + sigmoidf(p1)) * invG;
      const float pw  = __expf(p2) * aw;
      const float ph  = __expf(p3) * ah;
      const float x1p = clamp01(pcx - 0.5f * pw), y1p = clamp01(pcy - 0.5f * ph);
      const float x2p = clamp01(pcx + 0.5f * pw), y2p = clamp01(pcy + 0.5f * ph);

      const float tcx = ((float)gx + tl[0]) * invG;
      const float tcy = ((float)gy + tl[1]) * invG;
      const float tw  = __expf(tl[2]) * aw;
      const float th  = __expf(tl[3]) * ah;
      const float x1t = clamp01(tcx - 0.5f * tw), y1t = clamp01(tcy - 0.5f * th);
      const float x2t = clamp01(tcx + 0.5f * tw), y2t = clamp01(tcy + 0.5f * th);

      // CIoU loss element
      const float iw    = fmaxf(fminf(x2p, x2t) - fmaxf(x1p, x1t), 0.0f);
      const float ih    = fmaxf(fminf(y2p, y2t) - fmaxf(y1p, y1t), 0.0f);
      const float inter = iw * ih;
      const float areaP = fmaxf(x2p - x1p, 0.0f) * fmaxf(y2p - y1p, 0.0f);
      const float areaT = fmaxf(x2t - x1t, 0.0f) * fmaxf(y2t - y1t, 0.0f);
      const float uni   = areaP + areaT - inter + EPSF;
      const float iou   = inter / uni;
      const float dx    = (x1p + x2p - x1t - x2t) * 0.5f;
      const float dy    = (y1p + y2p - y1t - y2t) * 0.5f;
      const float rho2  = dx * dx + dy * dy;
      const float enw   = fmaxf(fmaxf(x2p, x2t) - fminf(x1p, x1t), EPSF);
      const float enh   = fmaxf(fmaxf(y2p, y2t) - fminf(y1p, y1t), EPSF);
      const float c2    = enw * enw + enh * enh;
      const float wp    = fmaxf(x2p - x1p, EPSF), hp = fmaxf(y2p - y1p, EPSF);
      const float wt    = fmaxf(x2t - x1t, EPSF), ht = fmaxf(y2t - y1t, EPSF);
      const float dv    = atanf(wt / ht) - atanf(wp / hp);
      const float v     = (4.0f / (PI_F * PI_F)) * dv * dv;
      const float alpha = v / (1.0f - iou + v + EPSF);
      const float ciou  = iou - rho2 / (c2 + EPSF) - alpha * v;
      acc_coord += 1.0f - ciou;

      float cls = 0.0f;
      #pragma unroll
      for (int c = 0; c < NCLS; ++c)
        cls += softplusf(-pb[(5 + c) * GG]);             // -log(sigmoid)
      acc_cls += cls;
    } else {
      acc_noobj += softplusf(p4);                        // -log1p(-sigmoid(p4))
    }

    __syncthreads();  // all reads of sTgt[bufSel] done before it is reused
  }

  // ---- wave32 reduction (5 quantities), then 8-wave LDS combine
  float s0 = acc_nobj, s1 = acc_coord, s2 = acc_objb, s3 = acc_noobj, s4 = acc_cls;
  #pragma unroll
  for (int off = 16; off > 0; off >>= 1) {
    s0 += __shfl_down(s0, off, 32);
    s1 += __shfl_down(s1, off, 32);
    s2 += __shfl_down(s2, off, 32);
    s3 += __shfl_down(s3, off, 32);
    s4 += __shfl_down(s4, off, 32);
  }
  const int wave = tid >> 5, lane = tid & 31;
  if (lane == 0) {
    sRed[wave][0] = s0; sRed[wave][1] = s1; sRed[wave][2] = s2;
    sRed[wave][3] = s3; sRed[wave][4] = s4;
  }
  __syncthreads();
  if (tid == 0) {
    float r[5] = {0.0f, 0.0f, 0.0f, 0.0f, 0.0f};
    for (int w = 0; w < 8; ++w)
      for (int q = 0; q < 5; ++q) r[q] += sRed[w][q];
    for (int q = 0; q < 5; ++q) ws[(size_t)blockIdx.x * 5 + q] = r[q];
  }
}

// ---------------------------------------------------------------------------
// Pass 2: deterministic final reduce of block partials + loss combination.
// ---------------------------------------------------------------------------
__global__ void __launch_bounds__(256) yolo_finalize(
    const float* __restrict__ ws, float* __restrict__ out, int nblocks)
{
  __shared__ float sRed[8][5];
  float s[5] = {0.0f, 0.0f, 0.0f, 0.0f, 0.0f};
  for (int i = threadIdx.x; i < nblocks; i += 256)
    for (int q = 0; q < 5; ++q) s[q] += ws[(size_t)i * 5 + q];

  #pragma unroll
  for (int off = 16; off > 0; off >>= 1)
    for (int q = 0; q < 5; ++q) s[q] += __shfl_down(s[q], off, 32);

  const int wave = threadIdx.x >> 5, lane = threadIdx.x & 31;
  if (lane == 0)
    for (int q = 0; q < 5; ++q) sRed[wave][q] = s[q];
  __syncthreads();

  if (threadIdx.x == 0) {
    float t[5] = {0.0f, 0.0f, 0.0f, 0.0f, 0.0f};
    for (int w = 0; w < 8; ++w)
      for (int q = 0; q < 5; ++q) t[q] += sRed[w][q];

    const float n_obj = t[0];
    const float n_noobj = (float)NCELLS - n_obj;
    const float coord_loss = (n_obj > 0.0f)   ? t[1] / fmaxf(n_obj, 1.0f)              : 0.0f;
    const float obj_loss   = (n_obj > 0.0f)   ? t[2] / fmaxf(n_obj, 1.0f)              : 0.0f;
    const float noobj_loss = (n_noobj > 0.0f) ? t[3] / fmaxf(n_noobj, 1.0f)            : 0.0f;
    const float class_loss = (n_obj > 0.0f)   ? t[4] / fmaxf(n_obj * (float)NCLS, 1.0f): 0.0f;
    out[0] = 10.0f * coord_loss + 1.0f * obj_loss + 0.5f * noobj_loss + class_loss;
  }
}

// ---------------------------------------------------------------------------
extern "C" void kernel_launch(void* const* d_in, const int* in_sizes, int n_in,
                              void* d_out, int out_size, void* d_ws, size_t ws_size,
                              hipStream_t stream) {
  const float* pred    = (const float*)d_in[0];
  const float* tgt     = (const float*)d_in[1];
  const float* anchors = (const float*)d_in[2];
  float* out = (float*)d_out;
  float* ws  = (float*)d_ws;

  int nb = 1024;
  if (nb > NCHUNKS) nb = NCHUNKS;
  size_t need = (size_t)nb * 5 * sizeof(float);
  if (ws_size < need) nb = (int)(ws_size / (5 * sizeof(float)));
  if (nb < 1) nb = 1;

  yolo_ciou_partials<<<nb, 256, 0, stream>>>(pred, tgt, anchors, ws);
  yolo_finalize<<<1, 256, 0, stream>>>(ws, out, nb);
}